// LIFSpike_58231166599336
// MI455X (gfx1250) — compile-verified
//
#include <hip/hip_runtime.h>

// LIF spike forward scan, MI455X (gfx1250).
// [T=4, B=64, C=128, H=32, W=32] fp32. Memory-bound streaming scan:
// 268 MB @ 23.3 TB/s ~= 11.5 us floor, ~0.5 GFLOP -> no matrix math, no WMMA.
//
// Data movement uses the CDNA5 async global->LDS path in a double-buffered
// ASYNCcnt pipeline:
//   - each stage issues exactly 4 GLOBAL_LOAD_ASYNC_TO_LDS_B128 (one per
//     t-plane, 16 B/lane, fully coalesced; planes are 32 MiB apart)
//   - while computing buffer k we prefetch buffer k+1, then
//     s_wait_asynccnt 0x4 releases buffer k (async loads complete in order)
//   - async loads park no VGPR destinations and give 8 b128 loads in flight
//     per wave, so HBM stays saturated even at reduced occupancy
// Compute: unrolled 4-step scan in registers; spikes written with
// non-temporal b128 stores (256 MB combined footprint > 192 MB L2, no reuse).

typedef __attribute__((ext_vector_type(4))) float v4f;

#ifndef LIF_T
#define LIF_T 4
#endif
#define BLOCK 256  // 8 wave32 per block

__global__ __launch_bounds__(BLOCK) void lif_scan_async(const float* __restrict__ x,
                                                        float* __restrict__ out,
                                                        long long plane_vec, // float4 per t-plane
                                                        int iters) {         // uniform trip count
    // [2 bufs][BLOCK threads][T planes][4 floats] = 32 KB
    __shared__ float stage[2 * BLOCK * LIF_T * 4];

    const v4f* __restrict__ xin = (const v4f*)x;
    v4f* __restrict__ o = (v4f*)out;

    const long long nthreads = (long long)gridDim.x * BLOCK;
    const long long maxi = plane_vec - 1;
    long long i = (long long)blockIdx.x * BLOCK + threadIdx.x;

    const float TAU = 0.99f;
    const float THRESH = 1.0f;

    // Issue 4 async b128 global->LDS loads for one staging buffer.
    // OOB lanes clamp the *address* (not the issue count) so ASYNCcnt
    // accounting stays identical across the wave.
    auto stage_tile = [&](int buf, long long idx) {
        long long j = (idx <= maxi) ? idx : maxi;
#pragma unroll
        for (int t = 0; t < LIF_T; ++t) {
            const v4f* g = xin + (long long)t * plane_vec + j;
            unsigned l = (unsigned)(unsigned long long)
                &stage[(((long long)buf * BLOCK + threadIdx.x) * LIF_T + t) * 4];
            asm volatile("global_load_async_to_lds_b128 %0, %1, off"
                         :
                         : "v"(l), "v"(g)
                         : "memory");
        }
    };

    int buf = 0;
    stage_tile(buf, i);

    for (int it = 0; it < iters; ++it) {
        const long long inext = i + nthreads;
        if (it + 1 < iters) {                 // scalar branch: iters is uniform
            stage_tile(buf ^ 1, inext);       // prefetch next tile (4 more async)
            asm volatile("s_wait_asynccnt 0x4" ::: "memory"); // current tile landed
        } else {
            asm volatile("s_wait_asynccnt 0x0" ::: "memory"); // drain
        }

        if (i <= maxi) {
            // Consume current buffer from LDS (ds_load_b128 x4).
            v4f xt[LIF_T];
#pragma unroll
            for (int t = 0; t < LIF_T; ++t)
                xt[t] = *(const v4f*)
                    &stage[(((long long)buf * BLOCK + threadIdx.x) * LIF_T + t) * 4];

            v4f mem = {0.0f, 0.0f, 0.0f, 0.0f};
#pragma unroll
            for (int t = 0; t < LIF_T; ++t) {
                v4f s;
#pragma unroll
                for (int k = 0; k < 4; ++k) {
                    float m = mem[k] * TAU + xt[t][k];  // leaky integrate
                    bool fire = (m >= THRESH);          // heaviside forward
                    s[k] = fire ? 1.0f : 0.0f;          // spike
                    mem[k] = fire ? 0.0f : m;           // hard reset
                }
                __builtin_nontemporal_store(s, o + (long long)t * plane_vec + i);
            }
        }

        buf ^= 1;
        i = inext;
    }
}

extern "C" void kernel_launch(void* const* d_in, const int* in_sizes, int n_in,
                              void* d_out, int out_size, void* d_ws, size_t ws_size,
                              hipStream_t stream) {
    (void)n_in; (void)d_ws; (void)ws_size; (void)out_size;

    const float* x = (const float*)d_in[0];
    float* out = (float*)d_out;

    const long long total = (long long)in_sizes[0];   // T*B*C*H*W = 33,554,432
    const long long plane = total / LIF_T;            // 8,388,608 (divisible by 4)
    const long long plane_vec = plane / 4;            // 2,097,152 float4 per plane

    // 2048 blocks x 256 thr = 524,288 threads -> exactly 4 pipelined
    // iterations per thread (generic clamp handles non-divisible sizes).
    const int grid = 2048;
    const long long nthreads = (long long)grid * BLOCK;
    const int iters = (int)((plane_vec + nthreads - 1) / nthreads);

    lif_scan_async<<<dim3(grid), dim3(BLOCK), 0, stream>>>(x, out, plane_vec, iters);
}